// LIFNeuron_66262755442978
// MI455X (gfx1250) — compile-verified
//
#include <hip/hip_runtime.h>
#include <stdint.h>

// LIF neuron scan: x[B=64, S=1024, F=512] f32 -> spikes, same shape.
//   m_t = 0.95*m_{t-1} + x_t ; s_t = (m_t > 0.5) ; m_t *= (1 - s_t)
// Memory-bound: 134 MB in + 134 MB out => ~11.5 us floor @ 23.3 TB/s HBM.
// 32768 independent (b,f) chains; vec2 per lane => 16384 lanes.
// CDNA5 path: double-buffered GLOBAL_LOAD_ASYNC_TO_LDS_B64 pipeline
// (ASYNCcnt), self-consumed per lane (no barriers), NT stores out.

#define LIF_B 64
#define LIF_S 1024
#define LIF_F 512
#define LIF_DECAY 0.95f
#define LIF_TH 0.5f

#define BLOCK 128             // 4 wave32s
#define CHUNK 32              // time steps per pipeline stage
#define NCHUNK (LIF_S / CHUNK)
#define CH2 (LIF_F / 2)       // 256 vec2-chains per batch row
#define NCHAIN (LIF_B * CH2)  // 16384
#define NBLK (NCHAIN / BLOCK) // 128 blocks
#define GSTRIDE (LIF_F / 2)   // v2f elements per time step = 256

typedef float v2f __attribute__((ext_vector_type(2)));

#if defined(__AMDGCN__) && __has_builtin(__builtin_amdgcn_global_load_async_to_lds_b64)
#define HAVE_ASYNC 1
#else
#define HAVE_ASYNC 0
#endif

#define STR_(x) #x
#define STR(x) STR_(x)

#if HAVE_ASYNC
#define WAIT_ASYNC(n) asm volatile("s_wait_asynccnt " STR(n) ::: "memory")
#define WAIT_DS(n)    asm volatile("s_wait_dscnt " STR(n) ::: "memory")

// Builtin's pointee type per hipcc diagnostic: int __attribute__((vector_size(8))).
// Must use the vector_size spelling (canonical match; ext_vector is a distinct kind).
typedef int b64vec __attribute__((vector_size(2 * sizeof(int))));
typedef __attribute__((address_space(1))) b64vec* glb_b64_ptr;   // "__device__" AS
typedef __attribute__((address_space(3))) b64vec* lds_b64_ptr;   // "__shared__" AS

// Generic LDS pointer: low 32 bits are the LDS offset (ISA aperture rule).
__device__ __forceinline__ lds_b64_ptr as_lds(const void* p) {
  return (lds_b64_ptr)(uint32_t)(uintptr_t)p;
}
__device__ __forceinline__ glb_b64_ptr as_glb(const void* p) {
  return (glb_b64_ptr)(uintptr_t)p;
}
#endif

__device__ __forceinline__ v2f lif_step(v2f& m, v2f xv) {
  m = m * LIF_DECAY + xv;     // v_fma per component
  v2f s;
#pragma unroll
  for (int i = 0; i < 2; ++i) {
    const bool fired = m[i] > LIF_TH;   // v_cmp
    s[i] = fired ? 1.0f : 0.0f;         // v_cndmask
    m[i] = fired ? 0.0f : m[i];         // v_cndmask (reset)
  }
  return s;
}

__global__ __launch_bounds__(BLOCK) void lif_scan_kernel(const float* __restrict__ x,
                                                         float* __restrict__ out) {
  const int tid   = threadIdx.x;
  const int chain = blockIdx.x * BLOCK + tid;            // vec2 chain id
  const int b     = chain / CH2;
  const int f2    = chain - b * CH2;
  const size_t base_f = (size_t)b * LIF_S * LIF_F + (size_t)f2 * 2;  // 8B aligned
  const v2f* __restrict__ gx = (const v2f*)(x + base_f);
  v2f* __restrict__ go       = (v2f*)(out + base_f);

  v2f m = {0.0f, 0.0f};

#if HAVE_ASYNC
  // [2 buffers][CHUNK t-steps][BLOCK lanes] of v2f = 64 KB.
  // Lane stride 8B -> ds_load_b64 covers all 64 banks conflict-free.
  __shared__ v2f buf[2 * CHUNK * BLOCK];

  // Prologue: issue chunk 0 (32 async b64 ops per wave).
#pragma unroll
  for (int tt = 0; tt < CHUNK; ++tt)
    __builtin_amdgcn_global_load_async_to_lds_b64(
        as_glb(gx + (size_t)tt * GSTRIDE),
        as_lds(&buf[tt * BLOCK + tid]), 0, 0);

  for (int k = 0; k < NCHUNK - 1; ++k) {
    const int slot  = k & 1;
    const int nslot = slot ^ 1;
    // WAR fence: prior ds reads of nslot (iter k-1) must retire before the
    // async engine overwrites it (async ops are unordered vs DS ops).
    WAIT_DS(0);
#pragma unroll
    for (int tt = 0; tt < CHUNK; ++tt)
      __builtin_amdgcn_global_load_async_to_lds_b64(
          as_glb(gx + (size_t)((k + 1) * CHUNK + tt) * GSTRIDE),
          as_lds(&buf[(nslot * CHUNK + tt) * BLOCK + tid]), 0, 0);
    // Async loads complete in order: <= CHUNK outstanding => chunk k landed.
    WAIT_ASYNC(CHUNK);
#pragma unroll
    for (int tt = 0; tt < CHUNK; ++tt) {
      v2f xv = buf[(slot * CHUNK + tt) * BLOCK + tid];
      v2f s  = lif_step(m, xv);
      __builtin_nontemporal_store(s, go + (size_t)(k * CHUNK + tt) * GSTRIDE);
    }
  }
  { // Epilogue: last chunk, nothing left to issue.
    const int k = NCHUNK - 1, slot = k & 1;
    WAIT_ASYNC(0);
#pragma unroll
    for (int tt = 0; tt < CHUNK; ++tt) {
      v2f xv = buf[(slot * CHUNK + tt) * BLOCK + tid];
      v2f s  = lif_step(m, xv);
      __builtin_nontemporal_store(s, go + (size_t)(k * CHUNK + tt) * GSTRIDE);
    }
  }
#else
  // Fallback: direct NT streaming loads, unrolled x8 for MLP, prefetch ahead
  // (__builtin_prefetch -> global_prefetch_b8 on gfx1250).
#pragma unroll 1
  for (int t = 0; t < LIF_S; t += 8) {
    if (t + 64 < LIF_S)
      __builtin_prefetch(gx + (size_t)(t + 64) * GSTRIDE, 0, 0);
    v2f xv[8];
#pragma unroll
    for (int u = 0; u < 8; ++u)
      xv[u] = __builtin_nontemporal_load(gx + (size_t)(t + u) * GSTRIDE);
#pragma unroll
    for (int u = 0; u < 8; ++u) {
      v2f s = lif_step(m, xv[u]);
      __builtin_nontemporal_store(s, go + (size_t)(t + u) * GSTRIDE);
    }
  }
#endif
}

extern "C" void kernel_launch(void* const* d_in, const int* in_sizes, int n_in,
                              void* d_out, int out_size, void* d_ws, size_t ws_size,
                              hipStream_t stream) {
  (void)in_sizes; (void)n_in; (void)d_ws; (void)ws_size; (void)out_size;
  const float* x = (const float*)d_in[0];
  float* out = (float*)d_out;
  lif_scan_kernel<<<dim3(NBLK), dim3(BLOCK), 0, stream>>>(x, out);
}